// HistogramRangeTracker_54279796686865
// MI455X (gfx1250) — compile-verified
//
#include <hip/hip_runtime.h>
#include <hip/hip_bf16.h>

// ---------------------------------------------------------------------------
// HistogramRangeTracker for MI455X (gfx1250, wave32)
//   pass 1: global min/max   (float4 loads, wave32 shfl reduce, u32-key atomics)
//   pass 2: 256-bin histogram (float4 loads, per-wave LDS sub-histograms, ds_add)
//   pass 3: cumsum via V_WMMA_F32_16X16X4_F32 (scan-as-matmul), threshold counts
// Workspace layout (u32): ws[0]=min-key, ws[1]=max-key, ws[2..257]=hist[256]
// ---------------------------------------------------------------------------

typedef float v2f __attribute__((ext_vector_type(2)));
typedef float v8f __attribute__((ext_vector_type(8)));

#define NBINS   256
#define SUBHIST 8      // one private LDS histogram per wave (256 thr = 8 waves)
#define HPAD    257    // stagger copies across LDS banks

// Monotone float -> u32 key (order-preserving for all non-NaN floats)
__device__ __forceinline__ unsigned f2key(float f) {
  unsigned b = __float_as_uint(f);
  return (b & 0x80000000u) ? ~b : (b | 0x80000000u);
}
__device__ __forceinline__ float key2f(unsigned k) {
  unsigned b = (k & 0x80000000u) ? (k ^ 0x80000000u) : ~k;
  return __uint_as_float(b);
}

__global__ void hrt_init_ws(unsigned* __restrict__ ws) {
  int t = threadIdx.x;                // launched with 256 threads
  if (t == 0) { ws[0] = 0xFFFFFFFFu; ws[1] = 0u; }
  ws[2 + t] = 0u;
}

__global__ void hrt_minmax(const float* __restrict__ in,
                           unsigned* __restrict__ ws, long long n) {
  long long gtid   = (long long)blockIdx.x * blockDim.x + threadIdx.x;
  long long stride = (long long)gridDim.x * blockDim.x;
  const float4* in4 = (const float4*)in;
  long long n4 = n >> 2;

  float lmin =  __builtin_inff();
  float lmax = -__builtin_inff();
  for (long long i = gtid; i < n4; i += stride) {
    __builtin_prefetch(&in4[i + stride], 0, 0);   // global_prefetch_b8
    float4 v = in4[i];                            // global_load_b128
    lmin = fminf(lmin, fminf(fminf(v.x, v.y), fminf(v.z, v.w)));
    lmax = fmaxf(lmax, fmaxf(fmaxf(v.x, v.y), fmaxf(v.z, v.w)));
  }
  if (gtid < (n & 3)) {                           // tail elements
    float x = in[(n4 << 2) + gtid];
    lmin = fminf(lmin, x); lmax = fmaxf(lmax, x);
  }

  // wave32 reduction (5 steps)
  #pragma unroll
  for (int m = 16; m >= 1; m >>= 1) {
    lmin = fminf(lmin, __shfl_xor(lmin, m, 32));
    lmax = fmaxf(lmax, __shfl_xor(lmax, m, 32));
  }

  __shared__ unsigned smin[SUBHIST], smax[SUBHIST];
  int wid  = threadIdx.x >> 5;
  int lane = threadIdx.x & 31;
  if (lane == 0) { smin[wid] = f2key(lmin); smax[wid] = f2key(lmax); }
  __syncthreads();
  if (threadIdx.x == 0) {
    unsigned kmin = smin[0], kmax = smax[0];
    #pragma unroll
    for (int w = 1; w < SUBHIST; ++w) {
      kmin = (smin[w] < kmin) ? smin[w] : kmin;
      kmax = (smax[w] > kmax) ? smax[w] : kmax;
    }
    atomicMin(&ws[0], kmin);
    atomicMax(&ws[1], kmax);
  }
}

__device__ __forceinline__ int binOf(float x, float gmin, float inv) {
  int b = (int)((x - gmin) * inv);
  b = (b < 0) ? 0 : b;
  return (b > NBINS - 1) ? (NBINS - 1) : b;
}

__global__ void hrt_hist(const float* __restrict__ in,
                         unsigned* __restrict__ ws, long long n) {
  __shared__ unsigned sh[SUBHIST * HPAD];
  for (int i = threadIdx.x; i < SUBHIST * HPAD; i += blockDim.x) sh[i] = 0u;
  __syncthreads();

  float gmin = key2f(ws[0]);
  float gmax = key2f(ws[1]);
  float inv  = (float)NBINS / (gmax - gmin);

  unsigned* my = &sh[(threadIdx.x >> 5) * HPAD];   // private per-wave histogram

  long long gtid   = (long long)blockIdx.x * blockDim.x + threadIdx.x;
  long long stride = (long long)gridDim.x * blockDim.x;
  const float4* in4 = (const float4*)in;
  long long n4 = n >> 2;

  for (long long i = gtid; i < n4; i += stride) {
    __builtin_prefetch(&in4[i + stride], 0, 0);
    float4 v = in4[i];
    atomicAdd(&my[binOf(v.x, gmin, inv)], 1u);     // ds_add_u32
    atomicAdd(&my[binOf(v.y, gmin, inv)], 1u);
    atomicAdd(&my[binOf(v.z, gmin, inv)], 1u);
    atomicAdd(&my[binOf(v.w, gmin, inv)], 1u);
  }
  if (gtid < (n & 3)) {
    float x = in[(n4 << 2) + gtid];
    atomicAdd(&my[binOf(x, gmin, inv)], 1u);
  }
  __syncthreads();

  // flush: one global atomic per bin per block
  for (int b = threadIdx.x; b < NBINS; b += blockDim.x) {
    unsigned s = 0;
    #pragma unroll
    for (int w = 0; w < SUBHIST; ++w) s += sh[w * HPAD + b];
    if (s) atomicAdd(&ws[2 + b], s);
  }
}

// One wave. cumsum of 256 bins via scan-as-matmul:
//   view hist as 16x16 row-major H; cum = H*U (+ per-row offsets),
//   H*U done as 16x16x16 = 4 chained V_WMMA_F32_16X16X4_F32.
__global__ void hrt_scan(const unsigned* __restrict__ ws,
                         float* __restrict__ out, long long n) {
  const unsigned* hist = ws + 2;
  int lane = threadIdx.x;          // 0..31, full wave (EXEC all 1s for WMMA)
  int m = lane & 15;               // A: row index / B: column index
  int h = lane >> 4;               // lane-half selects K sub-pair

  v8f acc = {};                    // C/D accumulator, 16x16 f32
  #pragma unroll
  for (int kk = 0; kk < 4; ++kk) {
    int k0 = 4 * kk + 2 * h;       // ISA 16x4 A-layout: half-waves hold K pairs
    v2f a, b;
    a.x = (float)hist[16 * m + k0];
    a.y = (float)hist[16 * m + k0 + 1];
    b.x = (k0     <= m) ? 1.0f : 0.0f;   // U upper-triangular (inclusive scan)
    b.y = (k0 + 1 <= m) ? 1.0f : 0.0f;
    acc = __builtin_amdgcn_wmma_f32_16x16x4_f32(
        /*neg_a=*/false, a, /*neg_b=*/false, b,
        /*c_mod=*/(short)0, acc, /*reuse_a=*/false, /*reuse_b=*/false);
  }

  // Row totals live at column 15 of each row in D-layout:
  //   row r -> VGPR (r&7), lane 15 + 16*(r>>3)
  float rt[16];
  #pragma unroll
  for (int r = 0; r < 16; ++r)
    rt[r] = __shfl(acc[r & 7], ((r >> 3) << 4) + 15, 32);

  float offr[16];
  float run = 0.0f;
  #pragma unroll
  for (int r = 0; r < 16; ++r) { offr[r] = run; run += rt[r]; }

  float total = (float)n;
  float lo = total * 0.005f;       // (1-0.99)/2
  float hi = total * 0.995f;       // (1+0.99)/2

  // cum is monotone => argmax(cum > t) == #{cum <= t}
  int cLo = 0, cHi = 0;
  #pragma unroll
  for (int v = 0; v < 8; ++v) {    // VGPR v holds row v (lanes 0-15) / v+8
    float o   = (h == 0) ? offr[v] : offr[v + 8];
    float cum = acc[v] + o;
    cLo += (cum <= lo);
    cHi += (cum <= hi);
  }
  #pragma unroll
  for (int s = 16; s >= 1; s >>= 1) {
    cLo += __shfl_xor(cLo, s, 32);
    cHi += __shfl_xor(cHi, s, 32);
  }

  if (lane == 0) {
    float gmin = key2f(ws[0]);
    float gmax = key2f(ws[1]);
    float step = (gmax - gmin) / (float)NBINS;   // linspace(min,max,257) step
    out[0] = gmin + (float)cLo * step;           // bin_edges[lower_idx]
    out[1] = gmin + (float)cHi * step;           // bin_edges[upper_idx]
  }
}

extern "C" void kernel_launch(void* const* d_in, const int* in_sizes, int n_in,
                              void* d_out, int out_size, void* d_ws, size_t ws_size,
                              hipStream_t stream) {
  const float* in = (const float*)d_in[0];
  long long n = (long long)in_sizes[0];
  unsigned* ws = (unsigned*)d_ws;
  float* out = (float*)d_out;

  const int threads = 256;          // 8 wave32 per block
  const int blocks  = 2048;         // grid-stride; ~0.5M threads streaming b128

  hrt_init_ws<<<1, 256, 0, stream>>>(ws);
  hrt_minmax <<<blocks, threads, 0, stream>>>(in, ws, n);
  hrt_hist   <<<blocks, threads, 0, stream>>>(in, ws, n);
  hrt_scan   <<<1, 32, 0, stream>>>(ws, out, n);
}